// ImportanceRenderer_45062796870298
// MI455X (gfx1250) — compile-verified
//
#include <hip/hip_runtime.h>
#include <hip/hip_bf16.h>
#include <math.h>
#include <stdint.h>

// ---------------------------------------------------------------------------
// MipRayMarcher2 forward for MI455X (gfx1250).
// Memory-bound streaming problem (~304 MB @ 23.3 TB/s ~= 13 us). Strategy:
//   - kernel A+B: two-stage global min/max of depths (sorted axis exploited)
//   - main kernel: 64 rays per 64-thread WG; coalesced ASYNC global->LDS
//     copies (GLOBAL_LOAD_ASYNC_TO_LDS_B128, ASYNCcnt), padded LDS rows to
//     kill bank conflicts, one ray per lane scan with v_exp/v_log hardware
//     transcendentals, weights staged in LDS and streamed out coalesced via
//     GLOBAL_STORE_ASYNC_FROM_LDS_B32.
// ---------------------------------------------------------------------------

#define NS 48          // samples per ray
#define NI 47          // intervals per ray
#define RAYS_PER_WG 64
#define CSTRIDE 148    // padded LDS row stride for colors (floats, 16B aligned)
#define ZSTRIDE 52     // padded LDS row stride for dens/depth (floats, 16B aligned)

typedef int v4i __attribute__((ext_vector_type(4)));
typedef __attribute__((address_space(1))) v4i gv4i_t;  // global int4
typedef __attribute__((address_space(3))) v4i lv4i_t;  // LDS int4
typedef __attribute__((address_space(1))) int gi32_t;  // global int
typedef __attribute__((address_space(3))) int li32_t;  // LDS int

__device__ __forceinline__ void async_copy_b128(const void* gsrc, void* ldsdst) {
#if __has_builtin(__builtin_amdgcn_global_load_async_to_lds_b128)
  __builtin_amdgcn_global_load_async_to_lds_b128(
      (gv4i_t*)(uintptr_t)gsrc, (lv4i_t*)ldsdst, 0, 0);
#else
  unsigned loff = (unsigned)(uintptr_t)(lv4i_t*)ldsdst;
  unsigned long long ga = (unsigned long long)(uintptr_t)gsrc;
  asm volatile("global_load_async_to_lds_b128 %0, %1, off"
               :: "v"(loff), "v"(ga) : "memory");
#endif
}

__device__ __forceinline__ void async_store_b32(void* gdst, const void* ldssrc) {
#if __has_builtin(__builtin_amdgcn_global_store_async_from_lds_b32)
  __builtin_amdgcn_global_store_async_from_lds_b32(
      (gi32_t*)(uintptr_t)gdst, (li32_t*)ldssrc, 0, 0);
#else
  unsigned loff = (unsigned)(uintptr_t)(li32_t*)ldssrc;
  unsigned long long ga = (unsigned long long)(uintptr_t)gdst;
  asm volatile("global_store_async_from_lds_b32 %0, %1, off"
               :: "v"(ga), "v"(loff) : "memory");
#endif
}

__device__ __forceinline__ void wait_async_zero() {
#if __has_builtin(__builtin_amdgcn_s_wait_asynccnt)
  __builtin_amdgcn_s_wait_asynccnt(0);
#else
  asm volatile("s_wait_asynccnt 0" ::: "memory");
#endif
}

// ---------------------------------------------------------------------------
// Stage 1: per-block partial min/max of depths (depths sorted along samples:
// min lives at sample 0, max at sample NS-1 of each ray).
// ---------------------------------------------------------------------------
#define RED_BLOCKS 64
#define RED_THREADS 256

__global__ __launch_bounds__(RED_THREADS)
void depth_minmax_partial(const float* __restrict__ depth,
                          float* __restrict__ part_min,
                          float* __restrict__ part_max,
                          int n_rays) {
  __shared__ float smin[RED_THREADS];
  __shared__ float smax[RED_THREADS];
  float lo = 1e30f, hi = -1e30f;
  for (int r = blockIdx.x * blockDim.x + threadIdx.x; r < n_rays;
       r += gridDim.x * blockDim.x) {
    lo = fminf(lo, depth[(size_t)r * NS]);
    hi = fmaxf(hi, depth[(size_t)r * NS + (NS - 1)]);
  }
  smin[threadIdx.x] = lo;
  smax[threadIdx.x] = hi;
  __syncthreads();
  for (int s = RED_THREADS / 2; s > 0; s >>= 1) {
    if (threadIdx.x < s) {
      smin[threadIdx.x] = fminf(smin[threadIdx.x], smin[threadIdx.x + s]);
      smax[threadIdx.x] = fmaxf(smax[threadIdx.x], smax[threadIdx.x + s]);
    }
    __syncthreads();
  }
  if (threadIdx.x == 0) {
    part_min[blockIdx.x] = smin[0];
    part_max[blockIdx.x] = smax[0];
  }
}

// Stage 2: single block folds the partials -> finals[0]=min, finals[1]=max.
__global__ __launch_bounds__(RED_BLOCKS)
void depth_minmax_final(const float* __restrict__ part_min,
                        const float* __restrict__ part_max,
                        float* __restrict__ finals) {
  __shared__ float smin[RED_BLOCKS];
  __shared__ float smax[RED_BLOCKS];
  smin[threadIdx.x] = part_min[threadIdx.x];
  smax[threadIdx.x] = part_max[threadIdx.x];
  __syncthreads();
  for (int s = RED_BLOCKS / 2; s > 0; s >>= 1) {
    if (threadIdx.x < s) {
      smin[threadIdx.x] = fminf(smin[threadIdx.x], smin[threadIdx.x + s]);
      smax[threadIdx.x] = fmaxf(smax[threadIdx.x], smax[threadIdx.x + s]);
    }
    __syncthreads();
  }
  if (threadIdx.x == 0) {
    finals[0] = smin[0];
    finals[1] = smax[0];
  }
}

// ---------------------------------------------------------------------------
// Main ray-march kernel: one ray per lane, 64 rays per workgroup.
// ---------------------------------------------------------------------------
__global__ __launch_bounds__(RAYS_PER_WG)
void raymarch_kernel(const float* __restrict__ colors,
                     const float* __restrict__ dens,
                     const float* __restrict__ depth,
                     const float* __restrict__ minmax,
                     float* __restrict__ out_rgb,
                     float* __restrict__ out_depth,
                     float* __restrict__ out_w) {
  __shared__ float sc[RAYS_PER_WG * CSTRIDE];  // colors, padded rows
  __shared__ float sd[RAYS_PER_WG * ZSTRIDE];  // densities -> then weights
  __shared__ float sz[RAYS_PER_WG * ZSTRIDE];  // depths, padded rows

  const int t = threadIdx.x;
  const int ray0 = blockIdx.x * RAYS_PER_WG;

  // ---- cooperative async copy: consecutive lanes fetch consecutive 16B
  //      global chunks (coalesced); LDS side lands in padded rows.
  {
    const char* g = (const char*)(colors + (size_t)ray0 * (NS * 3));
    char* l = (char*)sc;
#pragma unroll
    for (int k = 0; k < 36; ++k) {              // 64 rays * 36 chunks/row
      int gc = t + k * RAYS_PER_WG;             // 0..2303
      int row = gc / 36;
      int within = gc - row * 36;
      async_copy_b128(g + (size_t)gc * 16,
                      l + (size_t)row * (CSTRIDE * 4) + (size_t)within * 16);
    }
  }
  {
    const char* g = (const char*)(dens + (size_t)ray0 * NS);
    char* l = (char*)sd;
#pragma unroll
    for (int k = 0; k < 12; ++k) {              // 64 rays * 12 chunks/row
      int gc = t + k * RAYS_PER_WG;             // 0..767
      int row = gc / 12;
      int within = gc - row * 12;
      async_copy_b128(g + (size_t)gc * 16,
                      l + (size_t)row * (ZSTRIDE * 4) + (size_t)within * 16);
    }
  }
  {
    const char* g = (const char*)(depth + (size_t)ray0 * NS);
    char* l = (char*)sz;
#pragma unroll
    for (int k = 0; k < 12; ++k) {
      int gc = t + k * RAYS_PER_WG;
      int row = gc / 12;
      int within = gc - row * 12;
      async_copy_b128(g + (size_t)gc * 16,
                      l + (size_t)row * (ZSTRIDE * 4) + (size_t)within * 16);
    }
  }

  wait_async_zero();
  __syncthreads();

  const float dmin = minmax[0];
  const float dmax = minmax[1];

  const float* crow = sc + t * CSTRIDE;
  float*       drow = sd + t * ZSTRIDE;   // read density[i]; recycle slot i-1 for w
  const float* zrow = sz + t * ZSTRIDE;

  float pc0 = crow[0], pc1 = crow[1], pc2 = crow[2];
  float pd = drow[0];
  float pz = zrow[0];

  float T = 1.0f;
  float accR = 0.0f, accG = 0.0f, accB = 0.0f;
  float accZ = 0.0f, wsum = 0.0f;

#pragma unroll
  for (int i = 1; i < NS; ++i) {
    float c0 = crow[3 * i + 0];
    float c1 = crow[3 * i + 1];
    float c2 = crow[3 * i + 2];
    float d  = drow[i];
    float z  = zrow[i];

    float delta = z - pz;
    float dm = 0.5f * (d + pd) - 1.0f;
    // softplus via hardware v_exp/v_log; exact enough for |dm| << 17
    float sp = __logf(1.0f + __expf(dm));
    float alpha = 1.0f - __expf(-sp * delta);
    float w = alpha * T;

    drow[i - 1] = w;   // slot i-1 is dead; stage weight for coalesced flush

    accR += w * (c0 + pc0);
    accG += w * (c1 + pc1);
    accB += w * (c2 + pc2);
    accZ += w * (z + pz);
    wsum += w;

    T *= (1.0f - alpha + 1e-10f);

    pc0 = c0; pc1 = c1; pc2 = c2;
    pd = d; pz = z;
  }

  const int ray = ray0 + t;

  float cd = (0.5f * accZ) / wsum;
  if (isnan(cd)) cd = INFINITY;            // nan_to_num(nan=inf)
  cd = fminf(fmaxf(cd, dmin), dmax);       // clip collapses any inf too

  // (0.5*acc)*2 - 1 == acc - 1
  out_rgb[(size_t)ray * 3 + 0] = accR - 1.0f;
  out_rgb[(size_t)ray * 3 + 1] = accG - 1.0f;
  out_rgb[(size_t)ray * 3 + 2] = accB - 1.0f;
  out_depth[ray] = cd;

  // ---- flush weights: LDS padded rows -> contiguous global, coalesced b32
  __syncthreads();
  {
    char* gbase = (char*)(out_w + (size_t)ray0 * NI);  // 64*47 floats contiguous
    const char* lbase = (const char*)sd;
#pragma unroll
    for (int k = 0; k < NI; ++k) {               // 64*47 words total
      int idx = t + k * RAYS_PER_WG;             // 0..3007
      int row = idx / NI;
      int within = idx - row * NI;
      async_store_b32(gbase + (size_t)idx * 4,
                      lbase + (size_t)row * (ZSTRIDE * 4) + (size_t)within * 4);
    }
  }
  // S_ENDPGM performs an implicit wait-idle (drains ASYNCcnt) before retiring.
}

// ---------------------------------------------------------------------------
extern "C" void kernel_launch(void* const* d_in, const int* in_sizes, int n_in,
                              void* d_out, int out_size, void* d_ws,
                              size_t ws_size, hipStream_t stream) {
  (void)n_in; (void)out_size; (void)ws_size;

  const float* colors = (const float*)d_in[0];  // [B,R,48,3]
  const float* dens   = (const float*)d_in[1];  // [B,R,48,1]
  const float* depth  = (const float*)d_in[2];  // [B,R,48,1]
  float* out = (float*)d_out;

  const int n_rays = in_sizes[1] / NS;          // B*R = 262144

  float* ws      = (float*)d_ws;
  float* pmin    = ws;                          // [64]
  float* pmax    = ws + RED_BLOCKS;             // [64]
  float* finals  = ws + 2 * RED_BLOCKS;         // [2]

  depth_minmax_partial<<<RED_BLOCKS, RED_THREADS, 0, stream>>>(depth, pmin,
                                                              pmax, n_rays);
  depth_minmax_final<<<1, RED_BLOCKS, 0, stream>>>(pmin, pmax, finals);

  float* out_rgb   = out;                          // [n_rays,3]
  float* out_depth = out + (size_t)n_rays * 3;     // [n_rays]
  float* out_w     = out + (size_t)n_rays * 4;     // [n_rays,47]

  raymarch_kernel<<<n_rays / RAYS_PER_WG, RAYS_PER_WG, 0, stream>>>(
      colors, dens, depth, finals, out_rgb, out_depth, out_w);
}